// P2RLossV8_47115791237315
// MI455X (gfx1250) — compile-verified
//
#include <hip/hip_runtime.h>
#include <hip/hip_bf16.h>

typedef __attribute__((ext_vector_type(2))) float v2f;
typedef __attribute__((ext_vector_type(8))) float v8f;

#define BB   32
#define HH   768
#define WW   768
#define HWP  (HH * WW)          // 589824 pixels per batch
#define HW4  (HWP / 4)          // 147456 float4 per batch

#define NSL      32             // grid.y slices per batch
#define NTHREADS 256            // 8 waves
#define NWAVES   (NTHREADS / 32)
#define PER      (HW4 / NSL)            // 4608 float4 per block
#define WPER     (PER / NWAVES)         // 576 float4 per wave
#define WITER    (WPER / 32)            // 18 iterations, exact
static_assert(NSL * PER == HW4, "slice split must be exact");
static_assert(NWAVES * WPER == PER, "wave split must be exact");
static_assert(32 * WITER == WPER, "iter split must be exact");

#define ACC_PRED   0            // ws[0..31]  : per-batch pred sums
#define ACC_TGT    32           // ws[32..63] : per-batch target sums
#define ACC_MSE    64           // ws[64..95] : per-batch sum of squared diffs
#define ACC_FLOATS 128          // target map starts here (16-float aligned)

#define WMMA_SUM(cacc, a0, a1, ones)                                                        \
    cacc = __builtin_amdgcn_wmma_f32_16x16x4_f32(false, a0, false, ones, (short)0, cacc,    \
                                                 false, false);                             \
    cacc = __builtin_amdgcn_wmma_f32_16x16x4_f32(false, a1, false, ones, (short)0, cacc,    \
                                                 false, false);

// ---------------- kernel 1: zero accumulators + target map ----------------
__global__ void k_zero(float* ws, long total_floats) {
    long i      = (long)blockIdx.x * blockDim.x + threadIdx.x;
    long stride = (long)gridDim.x * blockDim.x;
    float4* p   = (float4*)ws;
    long n4     = total_floats >> 2;
    float4 z    = make_float4(0.f, 0.f, 0.f, 0.f);
    for (long j = i; j < n4; j += stride) p[j] = z;
}

// ---------------- kernel 2: Gaussian splat scatter + target sums ----------
__global__ void k_scatter(const int* __restrict__ pts, const int* __restrict__ cell_area,
                          float* ws, int N) {
    int t = blockIdx.x * blockDim.x + threadIdx.x;
    if (t >= BB * N) return;
    int b = t / N, n = t - b * N;
    float ca = (float)cell_area[0];
    int x = pts[(long)(b * N + n) * 2 + 0];
    int y = pts[(long)(b * N + n) * 2 + 1];
    // clip(float(x)/ca, 0, W-1) then truncate toward zero (matches jnp astype(int32))
    int px = (int)fminf(fmaxf((float)x / ca, 0.f), (float)(WW - 1));
    int py = (int)fminf(fmaxf((float)y / ca, 0.f), (float)(HH - 1));
    float* tgt = ws + ACC_FLOATS + (long)b * HWP;
    float wsum = 0.f;
    #pragma unroll
    for (int dy = -2; dy <= 2; ++dy) {
        #pragma unroll
        for (int dx = -2; dx <= 2; ++dx) {
            int ny = py + dy, nx = px + dx;
            if (ny >= 0 && ny < HH && nx >= 0 && nx < WW) {
                float w = expf(-0.5f * sqrtf((float)(dy * dy + dx * dx)));
                atomicAdd(tgt + (long)ny * WW + nx, w);
                wsum += w;
            }
        }
    }
    atomicAdd(ws + ACC_TGT + b, wsum);   // dense target sum == sum of valid weights
}

// ---------------- kernel 3: per-batch pred sum via fp32 WMMA --------------
// D = A x ones(4x16) + C accumulates every A element into each of 16 columns.
// Grand sum of the 16x16 C tile = 16 * true sum, in exact fp32 FMAs.
// Decomposition is exact (static_asserts) -> branch-free loop, EXEC all ones.
__global__ void __launch_bounds__(NTHREADS) k_predsum(const float* __restrict__ pred,
                                                      float* ws) {
    const int b    = blockIdx.x;
    const int wid  = threadIdx.x >> 5;
    const int lane = threadIdx.x & 31;
    const float4* src = (const float4*)(pred + (size_t)b * HWP)
                        + (size_t)blockIdx.y * PER + (size_t)wid * WPER + lane;

    v8f c = {0.f, 0.f, 0.f, 0.f, 0.f, 0.f, 0.f, 0.f};
    v2f ones; ones[0] = 1.f; ones[1] = 1.f;

    float4 v = src[0];                       // prologue load
    #pragma unroll 3
    for (int it = 0; it < WITER - 1; ++it) {
        int pf = (it + 9 < WITER) ? (it + 9) : (WITER - 1);
        __builtin_prefetch((const void*)(src + pf * 32), 0, 0);
        float4 vn = src[(it + 1) * 32];      // next load in flight before consume
        v2f a0; a0[0] = v.x; a0[1] = v.y;
        v2f a1; a1[0] = v.z; a1[1] = v.w;
        WMMA_SUM(c, a0, a1, ones);
        v = vn;
    }
    {   // epilogue
        v2f a0; a0[0] = v.x; a0[1] = v.y;
        v2f a1; a1[0] = v.z; a1[1] = v.w;
        WMMA_SUM(c, a0, a1, ones);
    }

    float p = c[0] + c[1] + c[2] + c[3] + c[4] + c[5] + c[6] + c[7];
    __shared__ float s_tot;
    if (threadIdx.x == 0) s_tot = 0.f;
    __syncthreads();
    atomicAdd(&s_tot, p);                    // ds_add_f32, one per lane
    __syncthreads();
    if (threadIdx.x == 0)
        atomicAdd(ws + ACC_PRED + b, s_tot * (1.0f / 16.0f));
}

// ---------------- kernel 4: normalized MSE via fp32 WMMA ------------------
__global__ void __launch_bounds__(NTHREADS) k_mse(const float* __restrict__ pred,
                                                  float* ws) {
    const int b    = blockIdx.x;
    const int wid  = threadIdx.x >> 5;
    const int lane = threadIdx.x & 31;
    const size_t off = (size_t)blockIdx.y * PER + (size_t)wid * WPER + lane;
    const float4* sp = (const float4*)(pred + (size_t)b * HWP) + off;
    const float4* st = (const float4*)(ws + ACC_FLOATS + (size_t)b * HWP) + off;

    const float invSp = 1.0f / (ws[ACC_PRED + b] + 1e-8f);
    const float invSt = 1.0f / ws[ACC_TGT + b];      // N>0 so sum>0 always

    v8f c = {0.f, 0.f, 0.f, 0.f, 0.f, 0.f, 0.f, 0.f};
    v2f ones; ones[0] = 1.f; ones[1] = 1.f;

    float4 a = sp[0];
    float4 t = st[0];
    #pragma unroll 3
    for (int it = 0; it < WITER - 1; ++it) {
        int pf = (it + 9 < WITER) ? (it + 9) : (WITER - 1);
        __builtin_prefetch((const void*)(sp + pf * 32), 0, 0);
        float4 an = sp[(it + 1) * 32];
        float4 tn = st[(it + 1) * 32];
        float d0 = a.x * invSp - t.x * invSt;
        float d1 = a.y * invSp - t.y * invSt;
        float d2 = a.z * invSp - t.z * invSt;
        float d3 = a.w * invSp - t.w * invSt;
        v2f a0; a0[0] = d0 * d0; a0[1] = d1 * d1;
        v2f a1; a1[0] = d2 * d2; a1[1] = d3 * d3;
        WMMA_SUM(c, a0, a1, ones);
        a = an; t = tn;
    }
    {   // epilogue
        float d0 = a.x * invSp - t.x * invSt;
        float d1 = a.y * invSp - t.y * invSt;
        float d2 = a.z * invSp - t.z * invSt;
        float d3 = a.w * invSp - t.w * invSt;
        v2f a0; a0[0] = d0 * d0; a0[1] = d1 * d1;
        v2f a1; a1[0] = d2 * d2; a1[1] = d3 * d3;
        WMMA_SUM(c, a0, a1, ones);
    }

    float p = c[0] + c[1] + c[2] + c[3] + c[4] + c[5] + c[6] + c[7];
    __shared__ float s_tot;
    if (threadIdx.x == 0) s_tot = 0.f;
    __syncthreads();
    atomicAdd(&s_tot, p);
    __syncthreads();
    if (threadIdx.x == 0)
        atomicAdd(ws + ACC_MSE + b, s_tot * (1.0f / 16.0f));
}

// ---------------- kernel 5: combine losses --------------------------------
__global__ void k_final(const float* __restrict__ log_scale, const int* __restrict__ cell_area,
                        const float* __restrict__ ws, float* out, int N) {
    __shared__ float s_cl, s_sl;
    if (threadIdx.x == 0) { s_cl = 0.f; s_sl = 0.f; }
    __syncthreads();
    if (threadIdx.x < BB) {
        float ca = (float)cell_area[0];
        atomicAdd(&s_cl, fabsf(ws[ACC_PRED + threadIdx.x] / ca - (float)N));
        atomicAdd(&s_sl, ws[ACC_MSE + threadIdx.x] * (1.0f / (float)HWP));
    }
    __syncthreads();
    if (threadIdx.x == 0) {
        float scale = expf(log_scale[0]);
        float scale_loss = fmaxf(8.0f - scale, 0.f) + fmaxf(scale - 64.0f, 0.f);
        out[0] = 2.0f * (s_cl * (1.0f / BB)) + 0.15f * (s_sl * (1.0f / BB))
               + 0.5f * scale_loss;
    }
}

extern "C" void kernel_launch(void* const* d_in, const int* in_sizes, int n_in,
                              void* d_out, int out_size, void* d_ws, size_t ws_size,
                              hipStream_t stream) {
    const float* pred      = (const float*)d_in[0];
    const int*   points    = (const int*)d_in[1];
    const float* log_scale = (const float*)d_in[2];
    const int*   cell_area = (const int*)d_in[3];
    float* out = (float*)d_out;
    float* ws  = (float*)d_ws;

    int N = in_sizes[1] / (BB * 2);                  // points: [B,N,2]
    long total_floats = ACC_FLOATS + (long)BB * HWP; // accumulators + target map

    k_zero<<<2048, 256, 0, stream>>>(ws, total_floats);
    k_scatter<<<(BB * N + 255) / 256, 256, 0, stream>>>(points, cell_area, ws, N);
    dim3 grid(BB, NSL);
    k_predsum<<<grid, NTHREADS, 0, stream>>>(pred, ws);
    k_mse<<<grid, NTHREADS, 0, stream>>>(pred, ws);
    k_final<<<1, 32, 0, stream>>>(log_scale, cell_area, ws, out, N);
}